// MultiheadAttention_35648228556929
// MI455X (gfx1250) — compile-verified
//
#include <hip/hip_runtime.h>
#include <math.h>

typedef float v2f __attribute__((ext_vector_type(2)));
typedef float v8f __attribute__((ext_vector_type(8)));

#define D_MODEL   1024
#define NUM_HEADS 16
#define D_HEAD    64
#define SEQ       2048
#define BATCH     2
#define MROWS     (BATCH * SEQ)          // 4096
#define SCALE     0.25f                  // 1/sqrt(NUM_HEADS), bug reproduced

__device__ __forceinline__ v8f wmma_f32(v2f a, v2f b, v8f c) {
  // D = A(16x4) * B(4x16) + C(16x16), fp32 WMMA, wave32
  return __builtin_amdgcn_wmma_f32_16x16x4_f32(false, a, false, b, (short)0, c,
                                               false, false);
}

// ---------------------------------------------------------------------------
// Y[m][n] = sum_k X[m][k] * W[n][k] + bias[n]
// One wave computes a 16x64 strip (4 N-tiles). K loop is split into an
// explicit load stage (20 b64 loads clause-batched) and a compute stage
// (16 WMMAs) per 16-deep K chunk so loadcnt waits stagger instead of
// stalling to zero before every WMMA.
//   mode 0: store flat       Y[M][D_MODEL]
//   mode 1: store split-head Y[B][H][S][dh]
// ---------------------------------------------------------------------------
__global__ __launch_bounds__(256)
void proj_gemm(const float* __restrict__ X, const float* __restrict__ W,
               const float* __restrict__ bias, float* __restrict__ Y, int mode)
{
  const int lane = threadIdx.x & 31;
  const int wave = threadIdx.x >> 5;
  const int tile = blockIdx.x * 8 + wave;
  const int NT4  = D_MODEL / 64;          // 16 strips along N
  const int mt   = tile / NT4;
  const int nt4  = tile % NT4;
  if (mt * 16 >= MROWS) return;           // wave-uniform guard
  const int half = lane >> 4;
  const int ln   = lane & 15;

  const float* xrow  = X + (size_t)(mt * 16 + ln) * D_MODEL;          // A: M = ln
  const float* wrow0 = W + (size_t)(nt4 * 64 + 0  + ln) * D_MODEL;    // B: N = ln
  const float* wrow1 = W + (size_t)(nt4 * 64 + 16 + ln) * D_MODEL;
  const float* wrow2 = W + (size_t)(nt4 * 64 + 32 + ln) * D_MODEL;
  const float* wrow3 = W + (size_t)(nt4 * 64 + 48 + ln) * D_MODEL;

  v8f acc0 = {}, acc1 = {}, acc2 = {}, acc3 = {};
#pragma unroll 1
  for (int k = 0; k < D_MODEL; k += 16) {
    v2f a[4], b0[4], b1[4], b2[4], b3[4];
#pragma unroll
    for (int u = 0; u < 4; ++u) {            // ---- load stage: 20 b64 loads ----
      const int kk = k + 4 * u + half * 2;
      a[u]  = v2f{ xrow[kk],  xrow[kk + 1]  };
      b0[u] = v2f{ wrow0[kk], wrow0[kk + 1] };
      b1[u] = v2f{ wrow1[kk], wrow1[kk + 1] };
      b2[u] = v2f{ wrow2[kk], wrow2[kk + 1] };
      b3[u] = v2f{ wrow3[kk], wrow3[kk + 1] };
    }
#pragma unroll
    for (int u = 0; u < 4; ++u) {            // ---- compute stage: 16 WMMAs ----
      acc0 = wmma_f32(a[u], b0[u], acc0);
      acc1 = wmma_f32(a[u], b1[u], acc1);
      acc2 = wmma_f32(a[u], b2[u], acc2);
      acc3 = wmma_f32(a[u], b3[u], acc3);
    }
  }

  v8f* accs[4] = { &acc0, &acc1, &acc2, &acc3 };
#pragma unroll
  for (int c = 0; c < 4; ++c) {
    const int colBase = nt4 * 64 + c * 16;
    const float bn = bias[colBase + ln];
    const v8f& acc = *accs[c];
#pragma unroll
    for (int r = 0; r < 8; ++r) {
      const int row = mt * 16 + r + 8 * half;   // global m
      const int col = colBase + ln;             // global n
      const float val = acc[r] + bn;
      if (mode == 0) {
        Y[(size_t)row * D_MODEL + col] = val;
      } else {
        const int b_ = row / SEQ;
        const int s_ = row % SEQ;
        const int h_ = col >> 6;
        const int d_ = col & 63;
        Y[(((size_t)(b_ * NUM_HEADS + h_) * SEQ) + s_) * D_HEAD + d_] = val;
      }
    }
  }
}

// ---------------------------------------------------------------------------
// Flash attention: one wave handles one (b, h, 16-row query tile).
// Qh/Kh/Vh are [B,H,S,dh]; output written merged as [B,S,D_MODEL].
// V-tile fragments are loaded BEFORE the softmax block so the shuffle/exp
// VALU work and the LDS P round-trip hide their latency.
// ---------------------------------------------------------------------------
__global__ __launch_bounds__(256)
void flash_attn(const float* __restrict__ Qh, const float* __restrict__ Kh,
                const float* __restrict__ Vh, float* __restrict__ Om)
{
  __shared__ float Pt[8][16 * 16];        // per-wave P-tile staging (C->A relayout)

  const int lane  = threadIdx.x & 31;
  const int wave  = threadIdx.x >> 5;
  const int gwave = blockIdx.x * 8 + wave;       // 4096 waves total
  const int qt    = gwave & 127;                  // 128 query tiles
  const int bh    = gwave >> 7;                   // 32 (b,h) pairs
  const int b     = bh >> 4;
  const int h     = bh & 15;
  const int half  = lane >> 4;
  const int ln    = lane & 15;

  const size_t headBase = (size_t)bh * SEQ * D_HEAD;
  const float* Q  = Qh + headBase;
  const float* Kp = Kh + headBase;
  const float* Vp = Vh + headBase;
  float* P = Pt[wave];

  // Preload Q tile in A-layout (16 chunks of 16x4), pre-scaled by SCALE.
  v2f qa[16];
  {
    const float* qrow = Q + (size_t)(qt * 16 + ln) * D_HEAD;
#pragma unroll
    for (int j = 0; j < 16; ++j) {
      const int d = 4 * j + half * 2;
      qa[j].x = qrow[d]     * SCALE;
      qa[j].y = qrow[d + 1] * SCALE;
    }
  }

  v8f o0 = {}, o1 = {}, o2 = {}, o3 = {};    // O tile 16x64 (4 column blocks)
  float mrow[8], lrow[8];
#pragma unroll
  for (int r = 0; r < 8; ++r) { mrow[r] = -INFINITY; lrow[r] = 0.f; }

  for (int kt = 0; kt < SEQ / 16; ++kt) {
    // ---- S = (Q*SCALE) @ K^T, 16 WMMAs over dh=64 ----
    v8f s = {};
    const float* krow = Kp + (size_t)(kt * 16 + ln) * D_HEAD;  // B: N(key) = ln
#pragma unroll
    for (int j = 0; j < 16; ++j) {
      const int d = 4 * j + half * 2;
      v2f bv = { krow[d], krow[d + 1] };
      s = wmma_f32(qa[j], bv, s);
    }

    // ---- prefetch V-tile B fragments (independent of softmax) ----
    const float* vbase = Vp + (size_t)(kt * 16) * D_HEAD;
    v2f vb0[4], vb1[4], vb2[4], vb3[4];
#pragma unroll
    for (int j = 0; j < 4; ++j) {
      const int kk = 4 * j + half * 2;
      const float* v0 = vbase + (size_t)kk * D_HEAD + ln;
      const float* v1 = v0 + D_HEAD;
      vb0[j] = v2f{ v0[0],  v1[0]  };
      vb1[j] = v2f{ v0[16], v1[16] };
      vb2[j] = v2f{ v0[32], v1[32] };
      vb3[j] = v2f{ v0[48], v1[48] };
    }

    // ---- online softmax; vgpr r of lane -> row (r + 8*half), col ln ----
#pragma unroll
    for (int r = 0; r < 8; ++r) {
      float v = s[r];
      float mx = v;
      mx = fmaxf(mx, __shfl_xor(mx, 1));
      mx = fmaxf(mx, __shfl_xor(mx, 2));
      mx = fmaxf(mx, __shfl_xor(mx, 4));
      mx = fmaxf(mx, __shfl_xor(mx, 8));
      const float newm = fmaxf(mrow[r], mx);
      const float p    = __expf(v - newm);
      const float corr = __expf(mrow[r] - newm);
      float ps = p;
      ps += __shfl_xor(ps, 1);
      ps += __shfl_xor(ps, 2);
      ps += __shfl_xor(ps, 4);
      ps += __shfl_xor(ps, 8);
      lrow[r] = lrow[r] * corr + ps;
      mrow[r] = newm;
      o0[r] *= corr; o1[r] *= corr; o2[r] *= corr; o3[r] *= corr;
      P[(r + 8 * half) * 16 + ln] = p;   // C-layout -> row-major LDS
    }

    // ---- O += P @ V  (A from LDS in A-layout; B from registers) ----
#pragma unroll
    for (int j = 0; j < 4; ++j) {
      const int kk = 4 * j + half * 2;
      v2f a = { P[ln * 16 + kk], P[ln * 16 + kk + 1] };   // A: M = ln
      o0 = wmma_f32(a, vb0[j], o0);
      o1 = wmma_f32(a, vb1[j], o1);
      o2 = wmma_f32(a, vb2[j], o2);
      o3 = wmma_f32(a, vb3[j], o3);
    }
  }

  // ---- normalize and store merged [B,S,D_MODEL] ----
  float* outp = Om + ((size_t)b * SEQ + qt * 16) * D_MODEL + h * D_HEAD;
#pragma unroll
  for (int r = 0; r < 8; ++r) {
    const float inv = 1.0f / lrow[r];
    const size_t ro = (size_t)(r + 8 * half) * D_MODEL;
    outp[ro + 0  + ln] = o0[r] * inv;
    outp[ro + 16 + ln] = o1[r] * inv;
    outp[ro + 32 + ln] = o2[r] * inv;
    outp[ro + 48 + ln] = o3[r] * inv;
  }
}

// ---------------------------------------------------------------------------
extern "C" void kernel_launch(void* const* d_in, const int* in_sizes, int n_in,
                              void* d_out, int out_size, void* d_ws, size_t ws_size,
                              hipStream_t stream) {
  const float* q  = (const float*)d_in[0];
  const float* k  = (const float*)d_in[1];
  const float* v  = (const float*)d_in[2];
  const float* Wq = (const float*)d_in[3];
  const float* bq = (const float*)d_in[4];
  const float* Wk = (const float*)d_in[5];
  const float* bk = (const float*)d_in[6];
  const float* Wv = (const float*)d_in[7];
  const float* bv = (const float*)d_in[8];
  const float* Wo = (const float*)d_in[9];
  const float* bo = (const float*)d_in[10];
  float* out = (float*)d_out;

  const size_t headElems = (size_t)BATCH * NUM_HEADS * SEQ * D_HEAD;  // 4 Mi floats
  float* Qh = (float*)d_ws;
  float* Kh = Qh + headElems;
  float* Vh = Kh + headElems;
  float* Ao = Vh + headElems;   // merged attention output [B,S,D_MODEL]

  const int gemmBlocks = (MROWS / 16) * (D_MODEL / 64) / 8;    // 512 (16x64 strips)
  const int attnBlocks = (BATCH * NUM_HEADS * (SEQ / 16)) / 8; // 512

  proj_gemm<<<gemmBlocks, 256, 0, stream>>>(q, Wq, bq, Qh, 1);
  proj_gemm<<<gemmBlocks, 256, 0, stream>>>(k, Wk, bk, Kh, 1);
  proj_gemm<<<gemmBlocks, 256, 0, stream>>>(v, Wv, bv, Vh, 1);
  flash_attn<<<attnBlocks, 256, 0, stream>>>(Qh, Kh, Vh, Ao);
  proj_gemm<<<gemmBlocks, 256, 0, stream>>>(Ao, Wo, bo, out, 0);
}